// FasterRcnnPredictModel_34196529610780
// MI455X (gfx1250) — compile-verified
//
#include <hip/hip_runtime.h>
#include <math.h>

// ---------------- problem constants ----------------
#define N_PROP   2000
#define N_PAD    2048          // padded proposal count (power of two)
#define WORDS    64            // 2048 bits per suppression row
#define N_CLS    80
#define MAX_PC   100
#define MAX_PI   300
#define IOU_THR  0.7f
#define PAD_KEY  0x007FFFFFu   // monotone key of -inf
#define TOT_CAND 8192          // 8000 candidates padded to power of two

typedef __attribute__((ext_vector_type(2))) float v2f;
typedef __attribute__((ext_vector_type(8))) float v8f;
typedef unsigned long long u64;

// monotone float->uint mapping (order-preserving for all floats)
__device__ __forceinline__ unsigned fkey(float f) {
  unsigned u = __float_as_uint(f);
  return (u & 0x80000000u) ? ~u : (u | 0x80000000u);
}

// ---------------- K1: decode boxes + areas ----------------
__global__ void k_decode(const float* __restrict__ deltas,
                         const float* __restrict__ props,
                         float* __restrict__ boxes,   // [N_PAD][4]
                         float* __restrict__ area) {  // [N_PAD]
  int i = blockIdx.x * blockDim.x + threadIdx.x;
  if (i >= N_PAD) return;
  float b0 = 0.f, b1 = 0.f, b2 = 0.f, b3 = 0.f;
  if (i < N_PROP) {
    float x1 = props[i*4+0], y1 = props[i*4+1];
    float x2 = props[i*4+2], y2 = props[i*4+3];
    float w  = x2 - x1 + 1.0f, h = y2 - y1 + 1.0f;
    float cx = x1 + 0.5f*w,   cy = y1 + 0.5f*h;
    float dx = deltas[i*4+0], dy = deltas[i*4+1];
    float dw = deltas[i*4+2], dh = deltas[i*4+3];
    float pcx = dx*w + cx, pcy = dy*h + cy;
    float pw  = expf(dw)*w, ph = expf(dh)*h;
    b0 = pcx - 0.5f*pw; b1 = pcy - 0.5f*ph;
    b2 = pcx + 0.5f*pw; b3 = pcy + 0.5f*ph;
  }
  boxes[i*4+0] = b0; boxes[i*4+1] = b1;
  boxes[i*4+2] = b2; boxes[i*4+3] = b3;
  area[i] = fmaxf(b2 - b0, 0.f) * fmaxf(b3 - b1, 0.f);
}

// ---------------- K2: thresholded IoU bitmask via WMMA union tiles ----------
// grid = (125 row-tiles, 8), block = 256 (8 waves). Wave handles one 32-bit
// column word (two 16-wide WMMA tiles). union_base = area_i + area_j computed
// as a rank-2 GEMM with V_WMMA_F32_16X16X4_F32: A=[area_i,1,0,0], B=[1;area_j;0;0].
// Threshold test is division-free: iou>thr  <=>  uni>0 && inter>thr*uni.
__global__ void k_iou_bits(const float4* __restrict__ boxes,
                           const float*  __restrict__ area,
                           unsigned* __restrict__ S) {  // [N_PROP][WORDS]
  __shared__ float4 rowBox[16];
  const int tid  = threadIdx.x;
  const int lane = tid & 31;
  const int wave = tid >> 5;
  const int row0 = blockIdx.x * 16;
  const int colWord = blockIdx.y * 8 + wave;
  const int col0 = colWord * 32;

  if (tid < 16) rowBox[tid] = boxes[row0 + tid];
  __syncthreads();

  const bool lo16 = (lane < 16);
  // A fragment (16x4 f32): lanes 0-15 hold K=0,1; lanes 16-31 hold K=2,3 (zeros)
  v2f a;
  a.x = lo16 ? area[row0 + lane] : 0.0f;   // K=0 column = area_i
  a.y = lo16 ? 1.0f : 0.0f;                // K=1 column = 1

  unsigned bt[2][8];
#pragma unroll
  for (int t = 0; t < 2; ++t) {
    const int colBase = col0 + t * 16;
    // B fragment (4x16 f32): K=0 row = ones, K=1 row = area_j
    v2f b;
    b.x = lo16 ? 1.0f : 0.0f;
    b.y = lo16 ? area[colBase + lane] : 0.0f;
    v8f c = {0.f,0.f,0.f,0.f,0.f,0.f,0.f,0.f};
    // d[m][n] = area_i + area_j  (union before subtracting intersection)
    v8f d = __builtin_amdgcn_wmma_f32_16x16x4_f32(
        false, a, false, b, (short)0, c, false, false);

    float4 cb = boxes[colBase + (lane & 15)];
#pragma unroll
    for (int v = 0; v < 8; ++v) {
      int m = v + (lo16 ? 0 : 8);           // C/D fragment row mapping
      float4 rb = rowBox[m];
      float ix1 = fmaxf(rb.x, cb.x), iy1 = fmaxf(rb.y, cb.y);
      float ix2 = fminf(rb.z, cb.z), iy2 = fminf(rb.w, cb.w);
      float inter = fmaxf(ix2 - ix1, 0.f) * fmaxf(iy2 - iy1, 0.f);
      float uni = d[v] - inter;
      // division-free threshold: iou > THR  <=>  uni > 0 && inter > THR*uni
      bool pred = (uni > 0.f) && (inter > IOU_THR * uni);
      bt[t][v] = (unsigned)__ballot(pred);
    }
  }
  // ballot bits: lanes 0-15 -> row v (N=0..15), lanes 16-31 -> row v+8
  if (lane < 16) {
    unsigned lo, hi;
    if (lane < 8) { lo = bt[0][lane] & 0xFFFFu;  hi = bt[1][lane] & 0xFFFFu; }
    else          { lo = bt[0][lane-8] >> 16;    hi = bt[1][lane-8] >> 16;   }
    int row = row0 + lane;
    if (row < N_PROP) S[row * WORDS + colWord] = lo | (hi << 16);
  }
}

// ---------------- K3: per-class NMS (one block per class) ----------------
__global__ void k_nms(const float* __restrict__ score,   // [N_PROP][N_CLS]
                      const unsigned* __restrict__ S,    // [N_PROP][WORDS]
                      u64* __restrict__ cand) {          // [N_CLS*MAX_PC]
  __shared__ u64 srt[N_PAD];                             // 16 KB LDS
  const int c = blockIdx.x;
  const int tid = threadIdx.x;

  for (int i = tid; i < N_PAD; i += blockDim.x) {
    unsigned key = (i < N_PROP) ? fkey(score[i * N_CLS + c]) : PAD_KEY;
    srt[i] = ((u64)key << 32) | (unsigned)i;
  }
  __syncthreads();

  // bitonic sort, descending by key
  for (unsigned k = 2; k <= N_PAD; k <<= 1) {
    for (unsigned j = k >> 1; j; j >>= 1) {
      for (unsigned i = tid; i < N_PAD; i += blockDim.x) {
        unsigned ixj = i ^ j;
        if (ixj > i) {
          u64 A = srt[i], B = srt[ixj];
          bool desc = ((i & k) == 0);
          if (desc ? (A < B) : (A > B)) { srt[i] = B; srt[ixj] = A; }
        }
      }
      __syncthreads();
    }
  }

  // prefill this class's candidate slots with -inf pads
  for (int t = tid; t < MAX_PC; t += blockDim.x)
    cand[c * MAX_PC + t] = ((u64)PAD_KEY << 32) | (unsigned)(c << 16);
  __syncthreads();

  // wave-synchronous greedy scan: keep-set = 64 words, 2 per lane (registers)
  if (tid < 32) {
    const int lane = tid;
    const uint2* __restrict__ S2 = (const uint2*)S;   // 2 words/lane per row
    unsigned kw0 = 0u, kw1 = 0u;
    int kcount = 0;
    for (int i = 0; i < N_PAD && kcount < MAX_PC; ++i) {
      u64 e = srt[i];
      int oi = (int)(e & 0xFFFFFFFFu);
      if (oi >= N_PROP) break;   // pads sorted to the tail
      uint2 r = S2[oi * (WORDS / 2) + lane];          // global_load_b64
      bool sup = __any(((r.x & kw0) | (r.y & kw1)) != 0);
      if (!sup) {
        int w = oi >> 5;
        if ((w >> 1) == lane) {
          unsigned bit = 1u << (oi & 31);
          if (w & 1) kw1 |= bit; else kw0 |= bit;
        }
        if (lane == 0) {
          unsigned key = (unsigned)(e >> 32);
          cand[c * MAX_PC + kcount] =
              ((u64)key << 32) | (unsigned)((c << 16) | oi);
        }
        kcount++;
      }
    }
  }
}

// ---------------- K4: global top-300 + output ----------------
__global__ void k_topk(u64* __restrict__ cand,           // [TOT_CAND] in ws
                       const float4* __restrict__ boxes,
                       float* __restrict__ out) {
  const int tid = threadIdx.x;
  for (int i = N_CLS * MAX_PC + tid; i < TOT_CAND; i += blockDim.x)
    cand[i] = 0ull;
  __syncthreads();

  for (unsigned k = 2; k <= TOT_CAND; k <<= 1) {
    for (unsigned j = k >> 1; j; j >>= 1) {
      for (unsigned i = tid; i < TOT_CAND; i += blockDim.x) {
        unsigned ixj = i ^ j;
        if (ixj > i) {
          u64 A = cand[i], B = cand[ixj];
          bool desc = ((i & k) == 0);
          if (desc ? (A < B) : (A > B)) { cand[i] = B; cand[ixj] = A; }
        }
      }
      __syncthreads();
    }
  }

  if (tid < MAX_PI) {
    u64 e = cand[tid];
    unsigned payload = (unsigned)e;
    int cls = (int)(payload >> 16);
    int oi  = (int)(payload & 0xFFFFu);
    float4 b = boxes[oi];
    out[tid*4+0] = b.x; out[tid*4+1] = b.y;
    out[tid*4+2] = b.z; out[tid*4+3] = b.w;
    ((int*)out)[MAX_PI * 4 + tid] = cls;   // int32 class ids after 1200 floats
  }
}

// ---------------- launcher ----------------
extern "C" void kernel_launch(void* const* d_in, const int* in_sizes, int n_in,
                              void* d_out, int out_size, void* d_ws, size_t ws_size,
                              hipStream_t stream) {
  const float* deltas = (const float*)d_in[0];   // roi_bboxes_txtytwth [2000,4]
  const float* score  = (const float*)d_in[1];   // roi_score [2000,80]
  const float* props  = (const float*)d_in[2];   // rpn_proposals_bboxes [2000,4]

  // workspace layout (total 618,496 bytes)
  char* w = (char*)d_ws;
  float*    boxes = (float*)w;                        //  32768 B: [2048][4] f32
  float*    area  = (float*)(w + 32768);              //   8192 B: [2048] f32
  unsigned* S     = (unsigned*)(w + 40960);           // 512000 B: [2000][64] u32
  u64*      cand  = (u64*)(w + 40960 + 512000);       //  65536 B: [8192] u64

  k_decode  <<<dim3(N_PAD / 256),           256, 0, stream>>>(deltas, props, boxes, area);
  k_iou_bits<<<dim3(N_PROP / 16, WORDS / 8),256, 0, stream>>>((const float4*)boxes, area, S);
  k_nms     <<<dim3(N_CLS),                 256, 0, stream>>>(score, S, cand);
  k_topk    <<<dim3(1),                    1024, 0, stream>>>(cand, (const float4*)boxes,
                                                              (float*)d_out);
}